// ModulateConv2d_23192823399040
// MI455X (gfx1250) — compile-verified
//
#include <hip/hip_runtime.h>
#include <stdint.h>

// Problem constants (from reference): B=4, C=64, Hin=Win=258, Hout=Wout=256, K=3
// out[b,c,h,w] = sum_k x[b,c,h+ki,w+kj] * mask[b,k,h,w] * W[c,k]
//
// Block: (b, 4-row strip, channel-half). LDS layout (dynamic, byte offsets):
//   maskS : 0      .. 36864   = 9 * 4 * 256 f32   (k-major, then row, then col)
//   wS    : 36864  .. 39168   = 64 * 9 f32
//   xS    : 39168  .. 51840   = 2 buffers * 6 rows * 264 f32 (row stride 1056 B)

typedef __attribute__((ext_vector_type(4))) unsigned int v4u;
typedef __attribute__((ext_vector_type(8))) int          v8i;
typedef __attribute__((ext_vector_type(4))) int          v4i;

#if __has_builtin(__builtin_amdgcn_s_wait_asynccnt)
#define WAIT_ASYNC0() __builtin_amdgcn_s_wait_asynccnt(0)
#else
#define WAIT_ASYNC0() asm volatile("s_wait_asynccnt 0x0" ::: "memory")
#endif

#if __has_builtin(__builtin_amdgcn_s_wait_tensorcnt)
#define WAIT_TENSOR0() __builtin_amdgcn_s_wait_tensorcnt(0)
#else
#define WAIT_TENSOR0() asm volatile("s_wait_tensorcnt 0x0" ::: "memory")
#endif

// Async global->LDS copies (GVS mode: SADDR base + 32-bit VGPR byte offset).
// vdst = LDS byte address VGPR; tracked with ASYNCcnt.
__device__ __forceinline__ void async_lds_b128(uint32_t ldsoff, const void* base, uint32_t goff) {
  asm volatile("global_load_async_to_lds_b128 %0, %1, %2 offset:0"
               :: "v"(ldsoff), "v"(goff), "s"(base) : "memory");
}
__device__ __forceinline__ void async_lds_b64(uint32_t ldsoff, const void* base, uint32_t goff) {
  asm volatile("global_load_async_to_lds_b64 %0, %1, %2 offset:0"
               :: "v"(ldsoff), "v"(goff), "s"(base) : "memory");
}

// Issue the 6x258 f32 x-tile for channel c into LDS buffer at bufoff.
// 129 lanes issue one b64 per row: lanes 0..127 cover cols 0..255, lane 128 cols 256..257.
__device__ __forceinline__ void issue_x_tile(const float* x, unsigned b, unsigned c,
                                             unsigned h0, uint32_t bufoff, int t) {
  if (t < 129) {
    const uint32_t colb = (uint32_t)t * 8u;
#pragma unroll
    for (int r = 0; r < 6; ++r) {
      const uint32_t goff = ((((b * 64u + c) * 258u + h0 + (unsigned)r) * 258u) << 2) + colb;
      async_lds_b64(bufoff + (uint32_t)r * 1056u + colb, x, goff);
    }
  }
}

extern "C" __global__ __launch_bounds__(256, 1)
void modconv_fused(const float* __restrict__ x, const float* __restrict__ mask,
                   const float* __restrict__ W, float* __restrict__ out) {
  extern __shared__ float smem[];
  const int t = threadIdx.x;
  const unsigned bid  = blockIdx.x;
  const unsigned cgrp = bid & 1u;          // channel half: 0 -> c 0..31, 1 -> c 32..63
  const unsigned tile = (bid >> 1) & 63u;  // 4-row strip index
  const unsigned b    = bid >> 7;          // batch
  const unsigned h0   = tile * 4u;
  const unsigned c0   = cgrp * 32u;

  const uint32_t MASK_OFF = 0u;
  const uint32_t W_OFF    = 36864u;
  const uint32_t X_OFF    = 39168u;

  // ---- Mask tile: one Tensor-Data-Mover descriptor, issued by wave 0 only ----
  // 2-D tile: dim0 = 4096 B contiguous (4 rows x 256 f32 of one k),
  //           dim1 = 9 (k), stride = 262144 B between k-planes.
#if __has_builtin(__builtin_amdgcn_tensor_load_to_lds)
  if (t < 32) {
    const uint64_t ga = (uint64_t)(uintptr_t)mask +
                        ((uint64_t)((b * 9u) * 65536u + h0 * 256u) << 2);
    v4u g0;
    g0.x = 1u;                                            // count=1, user mode, no gather
    g0.y = MASK_OFF;                                      // lds_addr
    g0.z = (uint32_t)ga;                                  // global_addr[31:0]
    g0.w = (uint32_t)((ga >> 32) & 0x01FFFFFFull)         // global_addr[56:32]
           | (2u << 30);                                  // type=2 ("image")
    v8i g1 = {
      0,                                                  // wg_mask=0, data_size=1B
      (int)(4096u << 16),                                 // tensor_dim0 lo16 -> bits 63:48
      (int)(9u << 16),                                    // tensor_dim0 hi=0, tensor_dim1 lo16=9
      (int)(4096u << 16),                                 // tensor_dim1 hi=0, tile_dim0=4096
      9,                                                  // tile_dim1=9, tile_dim2=0
      (int)262144,                                        // tensor_dim0_stride lo32
      0, 0                                                // stride hi / dim1_stride (unused)
    };
    v4i g2 = {0, 0, 0, 0};
    v4i g3 = {0, 0, 0, 0};
    v8i g4 = {0, 0, 0, 0, 0, 0, 0, 0};                    // extra operand on clang-23 lane
    __builtin_amdgcn_tensor_load_to_lds(g0, g1, g2, g3, g4, 0);
  }
#else
  // Fallback: per-lane async b128 (9 chunks x 4 KB, one b128 per lane per chunk)
  for (int k = 0; k < 9; ++k) {
    const uint32_t goff = (((b * 9u + (unsigned)k) * 65536u + h0 * 256u) << 2) + (uint32_t)t * 16u;
    async_lds_b128(MASK_OFF + (uint32_t)k * 4096u + (uint32_t)t * 16u, mask, goff);
  }
#endif

  // W table (64x9 f32 = 2304 B) via async b128
  if (t < 144) async_lds_b128(W_OFF + (uint32_t)t * 16u, W, (uint32_t)t * 16u);

  // First x tile (channel c0) into buffer 0
  issue_x_tile(x, b, c0, h0, X_OFF, t);

#if __has_builtin(__builtin_amdgcn_tensor_load_to_lds)
  if (t < 32) WAIT_TENSOR0();
#endif
  WAIT_ASYNC0();
  __syncthreads();

  // Mask values for this lane's column are identical for every channel:
  // keep all 36 in registers for the whole channel loop.
  float mk[9][4];
#pragma unroll
  for (int k = 0; k < 9; ++k)
#pragma unroll
    for (int r = 0; r < 4; ++r)
      mk[k][r] = smem[(k * 4 + r) * 256 + t];

  const float* wsf = smem + (W_OFF >> 2);

#pragma unroll 1
  for (unsigned cc = 0; cc < 32u; ++cc) {
    const unsigned c   = c0 + cc;
    const uint32_t cur = X_OFF + (cc & 1u) * 6336u;
    const uint32_t nxt = X_OFF + ((cc + 1u) & 1u) * 6336u;

    if (cc) { WAIT_ASYNC0(); __syncthreads(); }           // buffer `cur` landed, `nxt` free
    if (cc + 1u < 32u) issue_x_tile(x, b, c + 1u, h0, nxt, t);  // overlap DMA with compute

    float w[9];
#pragma unroll
    for (int k = 0; k < 9; ++k) w[k] = wsf[c * 9u + k];   // uniform LDS broadcast

    const float* xb = smem + (cur >> 2);
    float xv[6][3];
#pragma unroll
    for (int r = 0; r < 6; ++r)
#pragma unroll
      for (int j = 0; j < 3; ++j)
        xv[r][j] = xb[r * 264 + t + j];

    size_t obase = ((((size_t)(b * 64u + c) << 8) + h0) << 8) + (unsigned)t;
#pragma unroll
    for (int r = 0; r < 4; ++r) {
      float a = 0.0f;
#pragma unroll
      for (int ki = 0; ki < 3; ++ki)
#pragma unroll
        for (int kj = 0; kj < 3; ++kj) {
          const int k = ki * 3 + kj;
          a = __builtin_fmaf(xv[r + ki][kj], w[k] * mk[k][r], a);
        }
      __builtin_nontemporal_store(a, &out[obase + ((size_t)r << 8)]);  // write-once -> NT
    }
  }
}

extern "C" void kernel_launch(void* const* d_in, const int* in_sizes, int n_in,
                              void* d_out, int out_size, void* d_ws, size_t ws_size,
                              hipStream_t stream) {
  (void)in_sizes; (void)n_in; (void)d_ws; (void)ws_size; (void)out_size;
  const float* x    = (const float*)d_in[0];
  const float* mask = (const float*)d_in[1];
  const float* W    = (const float*)d_in[2];
  float* out        = (float*)d_out;

  // grid: 4 batches * 64 four-row strips * 2 channel-halves = 512 blocks
  dim3 grid(512), block(256);
  const size_t shmem = 36864 + 2304 + 2 * 6336;  // 51840 B < 320 KB/WGP
  modconv_fused<<<grid, block, shmem, stream>>>(x, mask, W, out);
}